// RelationTransformerEncoderLayer_1769526526259
// MI455X (gfx1250) — compile-verified
//
#include <hip/hip_runtime.h>
#include <math.h>

// ---------------- problem constants (static in reference) ----------------
#define B_    4
#define L_    13294
#define C_    256
#define NH_   8
#define NL_   4
#define NP_   4
#define HD_   32          // C/NH
#define DFF_  1024
#define M_    (B_ * L_)   // 53176 rows
#define BLC_  ((size_t)M_ * C_)

// ---------------- WMMA f32 16x16x4 helpers ----------------
typedef float v2f __attribute__((ext_vector_type(2)));
typedef float v8f __attribute__((ext_vector_type(8)));

static __device__ __forceinline__ v8f wmma_f32(v2f a, v2f b, v8f c) {
    // (neg_a, A, neg_b, B, c_mod, C, reuse_a, reuse_b)
    return __builtin_amdgcn_wmma_f32_16x16x4_f32(false, a, false, b, (short)0, c,
                                                 false, false);
}

static __device__ __forceinline__ float wave_sum(float v) {
#pragma unroll
    for (int m = 16; m > 0; m >>= 1) v += __shfl_xor(v, m, 32);
    return v;
}

// ---------------- elementwise add: q = query + query_pos ----------------
__global__ void add_kernel(const float* __restrict__ a, const float* __restrict__ b,
                           float* __restrict__ out, size_t n) {
    size_t i = (size_t)blockIdx.x * blockDim.x + threadIdx.x;
    if (i < n) out[i] = a[i] + b[i];
}

// ---------------- generic WMMA GEMM: out = A[M,K] @ W[K,N] (+bias)(+resid) -------
// Each wave computes a 16x64 strip (4 N-tiles sharing one A fragment).
// Ragged M rows are handled by clamping the A address (garbage only pollutes
// D rows that are never stored) -> branch-free inner loop.
template <int N, int K>
__global__ __launch_bounds__(128) void gemm_wmma_kernel(
    const float* __restrict__ A, const float* __restrict__ W,
    const float* __restrict__ bias, const float* __restrict__ resid,
    float* __restrict__ out, int M) {
    const int wave = threadIdx.x >> 5;
    const int lane = threadIdx.x & 31;
    const int half = lane >> 4;
    const int lm   = lane & 15;

    const int ngroups = N >> 6;               // 64-column groups
    const int mtiles  = (M + 15) >> 4;
    const int job = blockIdx.x * (blockDim.x >> 5) + wave;
    if (job >= mtiles * ngroups) return;      // wave-uniform exit
    const int mtile = job / ngroups;
    const int ng    = job % ngroups;

    const int arow = mtile * 16 + lm;
    const int rowc = arow < M ? arow : (M - 1);     // clamp, no predication
    const float* Ap = A + (size_t)rowc * K + half * 2;
    const float* Wp = W + (size_t)(half * 2) * N + ng * 64 + lm;

    v8f acc0 = {}, acc1 = {}, acc2 = {}, acc3 = {};
#pragma unroll 4
    for (int k = 0; k < K; k += 4) {
        const v2f a = *(const v2f*)Ap;                 // global_load_b64
        v2f b0, b1, b2, b3;
        b0.x = Wp[0];      b0.y = Wp[N];
        b1.x = Wp[16];     b1.y = Wp[N + 16];
        b2.x = Wp[32];     b2.y = Wp[N + 32];
        b3.x = Wp[48];     b3.y = Wp[N + 48];
        acc0 = wmma_f32(a, b0, acc0);
        acc1 = wmma_f32(a, b1, acc1);
        acc2 = wmma_f32(a, b2, acc2);
        acc3 = wmma_f32(a, b3, acc3);
        Ap += 4;
        Wp += 4 * N;
    }

    const bool full = (mtile * 16 + 16) <= M;   // wave-uniform
#pragma unroll
    for (int t = 0; t < 4; t++) {
        v8f acc = (t == 0) ? acc0 : (t == 1) ? acc1 : (t == 2) ? acc2 : acc3;
        const int n  = ng * 64 + t * 16 + lm;
        const float bv = bias ? bias[n] : 0.f;
        if (full) {
#pragma unroll
            for (int i = 0; i < 8; i++) {
                const int m = mtile * 16 + half * 8 + i;
                float v = acc[i] + bv;
                if (resid) v += resid[(size_t)m * N + n];
                out[(size_t)m * N + n] = v;
            }
        } else {
#pragma unroll
            for (int i = 0; i < 8; i++) {
                const int m = mtile * 16 + half * 8 + i;
                if (m < M) {
                    float v = acc[i] + bv;
                    if (resid) v += resid[(size_t)m * N + n];
                    out[(size_t)m * N + n] = v;
                }
            }
        }
    }
}

// ---------------- softmax over NL*NP=16 per (row, head), in place ----------------
__global__ void softmax16_kernel(float* __restrict__ aw, int n) {
    int i = blockIdx.x * blockDim.x + threadIdx.x;
    if (i >= n) return;
    float* p = aw + (size_t)(i / NH_) * (NH_ * 16) + (i % NH_) * 16;
    float mx = -1e30f;
#pragma unroll
    for (int j = 0; j < 16; j++) mx = fmaxf(mx, p[j]);
    float s = 0.f;
#pragma unroll
    for (int j = 0; j < 16; j++) { float e = __expf(p[j] - mx); p[j] = e; s += e; }
    const float inv = 1.f / s;
#pragma unroll
    for (int j = 0; j < 16; j++) p[j] *= inv;
}

// ---------------- MSDA bilinear sampling ----------------
static __device__ __forceinline__ float corner(const float* __restrict__ vl,
                                               int yi, int xi, int H, int W, float w) {
    const bool valid = (yi >= 0) & (yi < H) & (xi >= 0) & (xi < W);
    const int yc = yi < 0 ? 0 : (yi > H - 1 ? H - 1 : yi);
    const int xc = xi < 0 ? 0 : (xi > W - 1 ? W - 1 : xi);
    const float g = vl[(size_t)(yc * W + xc) * C_];
    return valid ? w * g : 0.f;
}

// one wave per (row, head); lane = head-dim channel (HD == 32 == wave size)
__global__ __launch_bounds__(256) void msda_sample_kernel(
    const float* __restrict__ v, const float* __restrict__ off,
    const float* __restrict__ aw, const float* __restrict__ rp,
    float* __restrict__ out) {
    const int lvlH[4] = {100, 50, 25, 13};
    const int lvlW[4] = {100, 50, 25, 13};
    const int lvlS[4] = {0, 10000, 12500, 13125};

    const int wave = threadIdx.x >> 5;
    const int lane = threadIdx.x & 31;
    const int job  = blockIdx.x * 8 + wave;
    if (job >= M_ * NH_) return;              // wave-uniform
    const int row = job / NH_;
    const int h   = job % NH_;
    const int b   = row / L_;

    const float* offr = off + (size_t)row * (NH_ * NL_ * NP_ * 2) + h * (NL_ * NP_ * 2);
    const float* awr  = aw  + (size_t)row * (NH_ * NL_ * NP_)     + h * (NL_ * NP_);
    const float* rpr  = rp  + (size_t)row * (NL_ * 2);

    float acc = 0.f;
#pragma unroll
    for (int l = 0; l < NL_; l++) {
        const int H = lvlH[l], W = lvlW[l];
        const float* vl = v + ((size_t)b * L_ + lvlS[l]) * C_ + h * HD_ + lane;
        const float rx = rpr[l * 2 + 0];
        const float ry = rpr[l * 2 + 1];
#pragma unroll
        for (int p = 0; p < NP_; p++) {
            const float ox  = offr[(l * NP_ + p) * 2 + 0];
            const float oy  = offr[(l * NP_ + p) * 2 + 1];
            const float wgt = awr[l * NP_ + p];
            // loc*size - 0.5 = rp*size + off - 0.5   (off/norm * size == off)
            const float x = rx * (float)W + ox - 0.5f;
            const float y = ry * (float)H + oy - 0.5f;
            const float x0f = floorf(x), y0f = floorf(y);
            const float lx = x - x0f, ly = y - y0f;
            const int x0 = (int)x0f, y0 = (int)y0f;
            float s = corner(vl, y0,     x0,     H, W, (1.f - ly) * (1.f - lx))
                    + corner(vl, y0,     x0 + 1, H, W, (1.f - ly) * lx)
                    + corner(vl, y0 + 1, x0,     H, W, ly * (1.f - lx))
                    + corner(vl, y0 + 1, x0 + 1, H, W, ly * lx);
            acc += wgt * s;
        }
    }
    out[(size_t)row * C_ + h * HD_ + lane] = acc;
}

// ---------------- LayerNorm over C=256, one wave per row ----------------
__global__ __launch_bounds__(256) void layernorm_kernel(
    const float* __restrict__ x, const float* __restrict__ g,
    const float* __restrict__ bb, float* __restrict__ out, int M) {
    const int wave = threadIdx.x >> 5;
    const int lane = threadIdx.x & 31;
    const int row  = blockIdx.x * (blockDim.x >> 5) + wave;
    if (row >= M) return;
    const float* xr = x + (size_t)row * C_;
    float e[8], s = 0.f;
#pragma unroll
    for (int i = 0; i < 8; i++) { e[i] = xr[i * 32 + lane]; s += e[i]; }
    s = wave_sum(s);
    const float mu = s * (1.f / C_);
    float var = 0.f;
#pragma unroll
    for (int i = 0; i < 8; i++) { const float d = e[i] - mu; var += d * d; }
    var = wave_sum(var);
    const float rs = rsqrtf(var * (1.f / C_) + 1e-5f);
#pragma unroll
    for (int i = 0; i < 8; i++) {
        const int c = i * 32 + lane;
        out[(size_t)row * C_ + c] = (e[i] - mu) * rs * g[c] + bb[c];
    }
}

// ---------------- fused FFN + LN2: out = LN2(x + relu(xW1+b1)W2 + b2) ------------
// One block (8 waves) per 16-row tile. Hidden 16x1024 tile lives in LDS.
#define HSTR 1028   // padded stride (floats) to avoid LDS bank conflicts
__global__ __launch_bounds__(256) void ffn_fused_kernel(
    const float* __restrict__ X,
    const float* __restrict__ W1, const float* __restrict__ b1,
    const float* __restrict__ W2, const float* __restrict__ b2,
    const float* __restrict__ g2, const float* __restrict__ bt2,
    float* __restrict__ out, int M) {
    extern __shared__ float smem[];
    float* h = smem;                  // [16][HSTR]
    float* o = smem + 16 * HSTR;      // [16][256]

    const int wave = threadIdx.x >> 5;
    const int lane = threadIdx.x & 31;
    const int half = lane >> 4;
    const int lm   = lane & 15;
    const int mtile = blockIdx.x;

    const int arow = mtile * 16 + lm;
    const int rowc = arow < M ? arow : (M - 1);   // clamp, no predication
    const float* Xp = X + (size_t)rowc * C_ + half * 2;

    // ---- stage 1: h = relu(X_tile @ W1 + b1), each wave covers 128 hidden cols
    v8f acc[8];
#pragma unroll
    for (int t = 0; t < 8; t++) acc[t] = (v8f){};
    {
        const float* Wp = W1 + (size_t)(half * 2) * DFF_ + wave * 128 + lm;
#pragma unroll 4
        for (int k = 0; k < C_; k += 4) {
            const v2f a = *(const v2f*)Xp;           // global_load_b64
#pragma unroll
            for (int t = 0; t < 8; t++) {
                v2f b;
                b.x = Wp[t * 16];
                b.y = Wp[DFF_ + t * 16];
                acc[t] = wmma_f32(a, b, acc[t]);
            }
            Xp += 4;
            Wp += 4 * DFF_;
        }
    }
#pragma unroll
    for (int t = 0; t < 8; t++) {
        const int n = wave * 128 + t * 16 + lm;
        const float bv = b1[n];
#pragma unroll
        for (int i = 0; i < 8; i++) {
            const int m = half * 8 + i;
            h[m * HSTR + n] = fmaxf(acc[t][i] + bv, 0.f);
        }
    }
    __syncthreads();

    // ---- stage 2: o = h @ W2 + b2 + X_tile, each wave covers 32 out cols
    v8f c0 = {}, c1 = {};
    {
        const float* hp = h + lm * HSTR + half * 2;
        const float* Wp = W2 + (size_t)(half * 2) * C_ + wave * 32 + lm;
#pragma unroll 4
        for (int k = 0; k < DFF_; k += 4) {
            const v2f a = *(const v2f*)hp;           // ds_load_b64
            v2f b0, b1v;
            b0.x  = Wp[0];    b0.y  = Wp[C_];
            b1v.x = Wp[16];   b1v.y = Wp[C_ + 16];
            c0 = wmma_f32(a, b0, c0);
            c1 = wmma_f32(a, b1v, c1);
            hp += 4;
            Wp += 4 * C_;
        }
    }
#pragma unroll
    for (int t = 0; t < 2; t++) {
        v8f cc = t ? c1 : c0;
        const int n = wave * 32 + t * 16 + lm;
        const float bv = b2[n];
#pragma unroll
        for (int i = 0; i < 8; i++) {
            const int m = half * 8 + i;
            const int grow = mtile * 16 + m;
            const int gc = grow < M ? grow : (M - 1);
            const float res = X[(size_t)gc * C_ + n];   // clamped, store-guarded later
            o[m * C_ + n] = cc[i] + bv + res;
        }
    }
    __syncthreads();

    // ---- LN2: each wave normalizes 2 rows of the tile
#pragma unroll
    for (int r = 0; r < 2; r++) {
        const int m = wave * 2 + r;
        const int grow = mtile * 16 + m;
        float e[8], s = 0.f;
#pragma unroll
        for (int i = 0; i < 8; i++) { e[i] = o[m * C_ + i * 32 + lane]; s += e[i]; }
        s = wave_sum(s);
        const float mu = s * (1.f / C_);
        float var = 0.f;
#pragma unroll
        for (int i = 0; i < 8; i++) { const float d = e[i] - mu; var += d * d; }
        var = wave_sum(var);
        const float rs = rsqrtf(var * (1.f / C_) + 1e-5f);
        if (grow < M) {
#pragma unroll
            for (int i = 0; i < 8; i++) {
                const int c = i * 32 + lane;
                out[(size_t)grow * C_ + c] = (e[i] - mu) * rs * g2[c] + bt2[c];
            }
        }
    }
}

// ---------------- host orchestration ----------------
extern "C" void kernel_launch(void* const* d_in, const int* in_sizes, int n_in,
                              void* d_out, int out_size, void* d_ws, size_t ws_size,
                              hipStream_t stream) {
    const float* query = (const float*)d_in[0];
    const float* qpos  = (const float*)d_in[1];
    const float* rp    = (const float*)d_in[2];
    // d_in[3]/d_in[4]: spatial shapes / level starts (static constants, unused)
    const float* v_W   = (const float*)d_in[5];
    const float* v_b   = (const float*)d_in[6];
    const float* off_W = (const float*)d_in[7];
    const float* off_b = (const float*)d_in[8];
    const float* aw_W  = (const float*)d_in[9];
    const float* aw_b  = (const float*)d_in[10];
    const float* out_W = (const float*)d_in[11];
    const float* out_b = (const float*)d_in[12];
    const float* ln1_g = (const float*)d_in[13];
    const float* ln1_b = (const float*)d_in[14];
    const float* l1_W  = (const float*)d_in[15];
    const float* l1_b  = (const float*)d_in[16];
    const float* l2_W  = (const float*)d_in[17];
    const float* l2_b  = (const float*)d_in[18];
    const float* ln2_g = (const float*)d_in[19];
    const float* ln2_b = (const float*)d_in[20];
    float* outp = (float*)d_out;

    // workspace layout (floats)
    float* ws      = (float*)d_ws;
    float* q       = ws;                    // [M,256]  (reused as pre-LN1 buffer)
    float* v       = q   + BLC_;            // [M,256]
    float* off     = v   + BLC_;            // [M,256]  (reused as LN1 output x)
    float* aw      = off + BLC_;            // [M,128]
    float* sampled = aw  + (size_t)M_ * 128;// [M,256]
    float* xpre    = q;                     // out-proj + residual (q no longer needed)
    float* x       = off;                   // LN1 output (off no longer needed)

    const int mtiles = (M_ + 15) / 16;

    // 1. q = query + query_pos
    {
        const size_t n = BLC_;
        add_kernel<<<dim3((unsigned)((n + 255) / 256)), dim3(256), 0, stream>>>(
            query, qpos, q, n);
    }
    // 2-4. projections (WMMA GEMMs)
    {
        int jobs = mtiles * (C_ / 64);  // N=256 -> 4 groups
        dim3 g((jobs + 3) / 4), b(128);
        gemm_wmma_kernel<256, 256><<<g, b, 0, stream>>>(query, v_W, v_b, nullptr, v, M_);
        gemm_wmma_kernel<256, 256><<<g, b, 0, stream>>>(q, off_W, off_b, nullptr, off, M_);
        int jobs2 = mtiles * (128 / 64);
        gemm_wmma_kernel<128, 256><<<dim3((jobs2 + 3) / 4), b, 0, stream>>>(
            q, aw_W, aw_b, nullptr, aw, M_);
    }
    // 5. softmax over 16 per (row, head)
    {
        const int n = M_ * NH_;
        softmax16_kernel<<<dim3((n + 255) / 256), dim3(256), 0, stream>>>(aw, n);
    }
    // 6. bilinear sampling -> sampled [M, C] (head-major channels)
    {
        const int waves = M_ * NH_;
        msda_sample_kernel<<<dim3((waves + 7) / 8), dim3(256), 0, stream>>>(
            v, off, aw, rp, sampled);
    }
    // 7. output projection + residual(query) -> xpre
    {
        int jobs = mtiles * (C_ / 64);
        gemm_wmma_kernel<256, 256><<<dim3((jobs + 3) / 4), dim3(128), 0, stream>>>(
            sampled, out_W, out_b, query, xpre, M_);
    }
    // 8. LN1 -> x
    layernorm_kernel<<<dim3((M_ + 7) / 8), dim3(256), 0, stream>>>(xpre, ln1_g, ln1_b, x, M_);
    // 9. fused FFN + LN2 -> d_out
    {
        const size_t smem = (size_t)(16 * HSTR + 16 * C_) * sizeof(float); // ~82 KB
        ffn_fused_kernel<<<dim3(mtiles), dim3(256), smem, stream>>>(
            x, l1_W, l1_b, l2_W, l2_b, ln2_g, ln2_b, outp, M_);
    }
    (void)in_sizes; (void)n_in; (void)out_size; (void)ws_size;
}